// GeneralAttention_74148315398603
// MI455X (gfx1250) — compile-verified
//
#include <hip/hip_runtime.h>
#include <hip/hip_bf16.h>
#include <math.h>

// Problem constants (from reference)
#define BB      32
#define SS      8192
#define DD      256
#define TWO_D   512
#define NCHUNK  64            // chunks per batch; 8192/64 = 128 rows/chunk = 8 waves * 16 rows
#define PSTRIDE 520           // per-block partial: [M, L, pad, pad, acc[512]]

// ws float offsets
#define OFF_U     0                         // 512
#define OFF_CB    512                       // 32
#define OFF_ALIGN 1024                      // B*S = 262144
#define OFF_PART  (1024 + BB*SS)            // 2048 * PSTRIDE
#define OFF_ML    (OFF_PART + BB*NCHUNK*PSTRIDE)  // 64

typedef __attribute__((ext_vector_type(2))) float v2f;
typedef __attribute__((ext_vector_type(4))) float v4f;
typedef __attribute__((ext_vector_type(8))) float v8f;

#define NEG_INF (-__builtin_inff())

// ---------------------------------------------------------------------------
// Kernel A: u = Ua^T va  (512), cb[b] = (Wa^T va) . input[b]  (32)
// ---------------------------------------------------------------------------
__global__ void attn_precompute(const float* __restrict__ input,
                                const float* __restrict__ Wa,
                                const float* __restrict__ Ua,
                                const float* __restrict__ va,
                                float* __restrict__ ws) {
  __shared__ float sva[DD];
  __shared__ float sw[DD];
  const int tid = threadIdx.x;   // 256 threads
  sva[tid] = va[tid];
  __syncthreads();

  // u[c] = sum_e va[e] * Ua[e*512 + c]   (coalesced over c)
  for (int rep = 0; rep < 2; ++rep) {
    const int c = rep * 256 + tid;
    float s = 0.f;
    for (int e = 0; e < DD; ++e) s += sva[e] * Ua[(size_t)e * TWO_D + c];
    ws[OFF_U + c] = s;
  }

  // w[i] = sum_o va[o] * Wa[o*256 + i]
  {
    float s = 0.f;
    for (int o = 0; o < DD; ++o) s += sva[o] * Wa[(size_t)o * DD + tid];
    sw[tid] = s;
  }
  __syncthreads();

  // cb[b] = sum_i input[b,i] * w[i]
  if (tid < BB) {
    float s = 0.f;
    const float* ip = input + (size_t)tid * DD;
    for (int i = 0; i < DD; ++i) s += ip[i] * sw[i];
    ws[OFF_CB + tid] = s;
  }
}

// ---------------------------------------------------------------------------
// Kernel B: fused streaming pass.
// grid = (NCHUNK, B), block = 256 (8 waves). Each wave: one 16-row tile.
// Scores via V_WMMA_F32_16X16X4_F32 (B operand = u broadcast over N),
// then online-softmax weighted accumulation of the context rows.
// ---------------------------------------------------------------------------
__global__ void attn_main(const float* __restrict__ ctx,
                          const int*   __restrict__ lens,
                          float* __restrict__ ws) {
  __shared__ float lds_u[TWO_D];
  __shared__ float lds_ml[16];
  __shared__ float lds_acc[8 * TWO_D];

  const int tid   = threadIdx.x;
  const int wave  = tid >> 5;
  const int lane  = tid & 31;
  const int chunk = blockIdx.x;
  const int b     = blockIdx.y;

  // stage u into LDS
  lds_u[tid]       = ws[OFF_U + tid];
  lds_u[tid + 256] = ws[OFF_U + 256 + tid];
  __syncthreads();

  const float cb  = ws[OFF_CB + b];
  const int   len = lens[b];

  const int rowbase = chunk * 128 + wave * 16;
  const int mrow    = lane & 15;          // A-matrix row this lane supplies
  const int khalf   = (lane >> 4) * 2;    // lanes 16-31 supply K+2,K+3

  // ---- scores for 16 rows via WMMA: C[m][n] = sum_k A[m][k]*u[k] (all n) ----
  v8f c8 = {};
  const float* ap = ctx + ((size_t)b * SS + rowbase + mrow) * TWO_D + khalf;
  #pragma unroll 8
  for (int kb = 0; kb < TWO_D; kb += 4) {
    v2f a  = *reinterpret_cast<const v2f*>(ap + kb);
    v2f bf = *reinterpret_cast<const v2f*>(&lds_u[kb + khalf]);
    c8 = __builtin_amdgcn_wmma_f32_16x16x4_f32(
        /*neg_a=*/false, a, /*neg_b=*/false, bf,
        /*c_mod=*/(short)0, c8, /*reuse_a=*/false, /*reuse_b=*/false);
  }

  // lanes 0-15 hold scores[0..7] in c8[0..7]; lanes 16-31 hold scores[8..15]
  const int hi8 = (lane >> 4) << 3;
  float sc[8];
  #pragma unroll
  for (int j = 0; j < 8; ++j) {
    float v = c8[j] + cb;
    if (rowbase + j + hi8 >= len) v = NEG_INF;
    sc[j] = v;
  }

  // write align scores (one lane per half)
  if ((lane & 15) == 0) {
    float* al = ws + OFF_ALIGN + (size_t)b * SS + rowbase + hi8;
    #pragma unroll
    for (int j = 0; j < 8; ++j) al[j] = sc[j];
  }

  // tile max (uniform within halves; xor-16 combines the halves)
  float mloc = sc[0];
  #pragma unroll
  for (int j = 1; j < 8; ++j) mloc = fmaxf(mloc, sc[j]);
  float tmax = fmaxf(mloc, __shfl_xor(mloc, 16, 32));

  // weighted accumulation: lane holds acc elems c = q*128 + lane*4 + t
  v4f acc[4] = {{0,0,0,0},{0,0,0,0},{0,0,0,0},{0,0,0,0}};
  float lsum = 0.f;
  if (tmax > NEG_INF) {
    #pragma unroll
    for (int r = 0; r < 16; ++r) {
      float scr = __shfl(sc[r & 7], (r < 8) ? 0 : 16, 32);  // broadcast score r
      if (scr > NEG_INF) {
        float p = __expf(scr - tmax);
        lsum += p;
        const float* rowp = ctx + ((size_t)b * SS + rowbase + r) * TWO_D + lane * 4;
        #pragma unroll
        for (int q = 0; q < 4; ++q) {
          v4f v = *reinterpret_cast<const v4f*>(rowp + q * 128);
          acc[q] += p * v;
        }
      }
    }
  }

  // ---- block combine over 8 waves ----
  if (lane == 0) { lds_ml[wave * 2] = tmax; lds_ml[wave * 2 + 1] = lsum; }
  #pragma unroll
  for (int q = 0; q < 4; ++q)
    *reinterpret_cast<v4f*>(&lds_acc[wave * TWO_D + q * 128 + lane * 4]) = acc[q];
  __syncthreads();

  float M = NEG_INF;
  #pragma unroll
  for (int w = 0; w < 8; ++w) M = fmaxf(M, lds_ml[2 * w]);

  // hoisted per-wave rescale factors
  float scale[8];
  float L = 0.f;
  #pragma unroll
  for (int w = 0; w < 8; ++w) {
    float mw = lds_ml[2 * w];
    scale[w] = (mw > NEG_INF) ? __expf(mw - M) : 0.f;
    L += scale[w] * lds_ml[2 * w + 1];
  }

  float* pb = ws + OFF_PART + (size_t)(b * NCHUNK + chunk) * PSTRIDE;
  if (tid == 0) { pb[0] = M; pb[1] = L; }
  #pragma unroll
  for (int rep = 0; rep < 2; ++rep) {
    const int c = rep * 256 + tid;
    float s = 0.f;
    #pragma unroll
    for (int w = 0; w < 8; ++w) s += scale[w] * lds_acc[w * TWO_D + c];
    pb[4 + c] = s;
  }
}

// ---------------------------------------------------------------------------
// Kernel C: per-batch combine of NCHUNK partials -> attn_h, (M_b, L_b)
// grid = B, block = 256
// ---------------------------------------------------------------------------
__global__ void attn_combine(float* __restrict__ ws, float* __restrict__ out) {
  __shared__ float sM[NCHUNK];
  __shared__ float sL[NCHUNK];
  const int b = blockIdx.x, tid = threadIdx.x;
  if (tid < NCHUNK) {
    const float* pb = ws + OFF_PART + (size_t)(b * NCHUNK + tid) * PSTRIDE;
    sM[tid] = pb[0];
    sL[tid] = pb[1];
  }
  __syncthreads();

  float M = NEG_INF;
  for (int k = 0; k < NCHUNK; ++k) M = fmaxf(M, sM[k]);

  float L = 0.f;
  float scale[NCHUNK];
  for (int k = 0; k < NCHUNK; ++k) {
    scale[k] = (sM[k] > NEG_INF) ? __expf(sM[k] - M) : 0.f;
    L += scale[k] * sL[k];
  }

  for (int rep = 0; rep < 2; ++rep) {
    const int c = rep * 256 + tid;
    float s = 0.f;
    for (int k = 0; k < NCHUNK; ++k) {
      if (scale[k] != 0.f) {
        const float* pb = ws + OFF_PART + (size_t)(b * NCHUNK + k) * PSTRIDE;
        s += scale[k] * pb[4 + c];
      }
    }
    out[(size_t)b * TWO_D + c] = s / L;   // attn_h [1,B,2D]
  }
  if (tid == 0) { ws[OFF_ML + 2 * b] = M; ws[OFF_ML + 2 * b + 1] = L; }
}

// ---------------------------------------------------------------------------
// Kernel D: align_vectors = softmax probs, written after attn_h in d_out
// ---------------------------------------------------------------------------
__global__ void attn_probs(const float* __restrict__ ws, float* __restrict__ out) {
  const int idx = blockIdx.x * 256 + threadIdx.x;  // 0 .. B*S-1
  const int b = idx >> 13;                          // S = 8192
  const float M = ws[OFF_ML + 2 * b];
  const float L = ws[OFF_ML + 2 * b + 1];
  const float v = ws[OFF_ALIGN + idx];
  float p = 0.f;
  if (v > NEG_INF && M > NEG_INF) p = __expf(v - M) / L;
  out[(size_t)BB * TWO_D + idx] = p;                // align_vectors [1,B,S]
}

// ---------------------------------------------------------------------------
extern "C" void kernel_launch(void* const* d_in, const int* in_sizes, int n_in,
                              void* d_out, int out_size, void* d_ws, size_t ws_size,
                              hipStream_t stream) {
  const float* input = (const float*)d_in[0];
  const float* ctx   = (const float*)d_in[1];
  const float* Wa    = (const float*)d_in[2];
  const float* Ua    = (const float*)d_in[3];
  const float* va    = (const float*)d_in[4];
  const int*   lens  = (const int*)d_in[5];
  float* out = (float*)d_out;
  float* ws  = (float*)d_ws;

  attn_precompute<<<1, 256, 0, stream>>>(input, Wa, Ua, va, ws);
  attn_main<<<dim3(NCHUNK, BB), 256, 0, stream>>>(ctx, lens, ws);
  attn_combine<<<BB, 256, 0, stream>>>(ws, out);
  attn_probs<<<(BB * SS) / 256, 256, 0, stream>>>(ws, out);
}